// GATModel_71227737636872
// MI455X (gfx1250) — compile-verified
//
#include <hip/hip_runtime.h>
#include <hip/hip_bf16.h>
#include <math.h>

typedef __attribute__((ext_vector_type(2))) float    v2f;
typedef __attribute__((ext_vector_type(8))) float    v8f;
typedef __attribute__((ext_vector_type(4))) unsigned v4u;
typedef __attribute__((ext_vector_type(8))) int      v8i;
typedef __attribute__((ext_vector_type(4))) int      v4i;

#define NEG_SLOPE 0.2f

#if defined(__has_builtin)
#if __has_builtin(__builtin_amdgcn_tensor_load_to_lds) && \
    __has_builtin(__builtin_amdgcn_s_wait_tensorcnt)
#define USE_TDM 1
#endif
#endif
#ifndef USE_TDM
#define USE_TDM 0
#endif

constexpr unsigned ilog2c(unsigned v) { unsigned r = 0; while (v > 1) { v >>= 1; ++r; } return r; }

// ---------- order-preserving float<->uint mapping for atomicMax ----------
__device__ __forceinline__ unsigned mapf(float f) {
  unsigned u = __float_as_uint(f);
  return (u & 0x80000000u) ? ~u : (u | 0x80000000u);
}
__device__ __forceinline__ float unmapf(unsigned u) {
  return (u & 0x80000000u) ? __uint_as_float(u & 0x7FFFFFFFu)
                           : __uint_as_float(~u);
}

#if USE_TDM
// Issue one TDM 2D tile load: nRows x rowLenDw dwords, global row stride
// strideDw dwords, into LDS at byte offset ldsOff with +4-dword padding after
// each row (pad_interval encodes rowLenDw, pad_amount=3 -> 4 dwords).
__device__ __forceinline__ void tdm_load_2d(unsigned ldsOff, const float* gptr,
                                            unsigned rowLenDw, unsigned nRows,
                                            unsigned strideDw, unsigned padEnc)
{
  unsigned long long ga = (unsigned long long)(size_t)gptr;
  v4u g0;
  g0.x = 1u;                                       // count=1, user descriptor
  g0.y = ldsOff;                                   // lds_addr (bytes)
  g0.z = (unsigned)(ga & 0xFFFFFFFFu);             // global_addr[31:0]
  g0.w = (unsigned)((ga >> 32) & 0x01FFFFFFu)      // global_addr[56:32]
       | (2u << 30);                               // type = 2 ("image")
  // group1: data_size=2 (4B), pad_enable, pad_interval=padEnc, pad_amount=3 (4 dw)
  unsigned d0 = (2u << 16) | (1u << 20) | (padEnc << 22) | (3u << 25);
  v8i g1;
  g1[0] = (int)d0;
  g1[1] = (int)((rowLenDw & 0xFFFFu) << 16);                              // tensor_dim0[15:0]
  g1[2] = (int)(((rowLenDw >> 16) & 0xFFFFu) | ((nRows & 0xFFFFu) << 16));// td0 hi | td1 lo
  g1[3] = (int)(((nRows >> 16) & 0xFFFFu) | ((rowLenDw & 0xFFFFu) << 16));// td1 hi | tile_dim0
  g1[4] = (int)(nRows & 0xFFFFu);                                         // tile_dim1 (tile_dim2=0)
  g1[5] = (int)strideDw;                                                  // tensor_dim0_stride lo32
  g1[6] = 0;                                                              // s0 hi16 | s1 lo16
  g1[7] = 0;                                                              // s1 hi32
  v4i z4 = {0, 0, 0, 0};
  v8i z8 = {0, 0, 0, 0, 0, 0, 0, 0};
  // 6-arg form (clang-23 / therock-10.0): (g0, g1, g2, g3, g4, cpol)
  __builtin_amdgcn_tensor_load_to_lds(g0, g1, z4, z4, z8, 0);
}
#endif

// ---------- GEMM: C[M,NCOLS] = A[M,K] @ W[K,NCOLS] (+bias) via f32 WMMA ----------
// grid = (M/16, NCOLS/PANEL), block = (PANEL/16)*32 threads.
// Block stages the 16xK A tile and the K x PANEL W panel in LDS (via TDM when
// available), each wave owns one 16-col tile, two independent WMMA chains.
template<int K, int NCOLS, int PANEL>
__global__ void gemm_wmma_f32(const float* __restrict__ A,
                              const float* __restrict__ W,
                              const float* __restrict__ bias,
                              float* __restrict__ C)
{
  constexpr int SA = K + 4;       // padded A row stride (dwords)
  constexpr int SW = PANEL + 4;   // padded W row stride (dwords)
  __shared__ __align__(16) float As[16 * SA];
  __shared__ __align__(16) float Ws[K * SW];

  const int row0   = blockIdx.x * 16;
  const int panel0 = blockIdx.y * PANEL;
  const int tid    = threadIdx.x;

#if USE_TDM
  if (tid < 32) {
    tdm_load_2d((unsigned)(size_t)As, A + (size_t)row0 * K,
                (unsigned)K, 16u, (unsigned)K, ilog2c(K) - 1);
    tdm_load_2d((unsigned)(size_t)Ws, W + panel0,
                (unsigned)PANEL, (unsigned)K, (unsigned)NCOLS, ilog2c(PANEL) - 1);
    __builtin_amdgcn_s_wait_tensorcnt(0);
  }
#else
  for (int i = tid; i < 16 * K; i += blockDim.x) {
    int r = i / K, c = i % K;
    As[r * SA + c] = A[(size_t)(row0 + r) * K + c];
  }
  for (int i = tid; i < K * PANEL; i += blockDim.x) {
    int r = i / PANEL, c = i % PANEL;
    Ws[r * SW + c] = W[(size_t)r * NCOLS + panel0 + c];
  }
#endif
  __syncthreads();

  const int wave = tid >> 5;
  const int lane = tid & 31;
  const int half = lane >> 4;   // K sub-pair select
  const int l16  = lane & 15;
  const int colp = wave * 16 + l16;   // column within panel (wave == n-tile)

  v8f acc0 = {0.f, 0.f, 0.f, 0.f, 0.f, 0.f, 0.f, 0.f};
  v8f acc1 = {0.f, 0.f, 0.f, 0.f, 0.f, 0.f, 0.f, 0.f};
  constexpr int H = K / 2;
#pragma unroll
  for (int k = 0; k < H; k += 4) {
    // chain 0: k in [0, K/2)
    {
      v2f a = *(const v2f*)&As[l16 * SA + k + 2 * half];
      v2f b;
      b.x = Ws[(k + 2 * half + 0) * SW + colp];
      b.y = Ws[(k + 2 * half + 1) * SW + colp];
      acc0 = __builtin_amdgcn_wmma_f32_16x16x4_f32(
          false, a, false, b, (short)0, acc0, false, false);
    }
    // chain 1: k in [K/2, K)  (independent accumulator -> ILP)
    {
      const int k1 = k + H;
      v2f a = *(const v2f*)&As[l16 * SA + k1 + 2 * half];
      v2f b;
      b.x = Ws[(k1 + 2 * half + 0) * SW + colp];
      b.y = Ws[(k1 + 2 * half + 1) * SW + colp];
      acc1 = __builtin_amdgcn_wmma_f32_16x16x4_f32(
          false, a, false, b, (short)0, acc1, false, false);
    }
  }

  const int colg = panel0 + wave * 16 + l16;
  const float bv = bias ? bias[colg] : 0.0f;
#pragma unroll
  for (int v = 0; v < 8; ++v) {
    const int r = row0 + v + 8 * half;    // D layout: VGPR v -> row v+8*half, col l16
    C[(size_t)r * NCOLS + colg] = acc0[v] + acc1[v] + bv;
  }
}

// ---------- attention coefficients: alpha_{s,d}[n,h] = <h[n,h,:], a[h,:]> ----------
__global__ void attn_coef(const float* __restrict__ h,
                          const float* __restrict__ a_s,
                          const float* __restrict__ a_d,
                          float* __restrict__ out_s,
                          float* __restrict__ out_d,
                          int n4)
{
  int idx = blockIdx.x * blockDim.x + threadIdx.x;
  if (idx >= n4) return;
  const int hd = idx & 3;
  const float4* hp = (const float4*)(h + (size_t)idx * 32);
  const float4* sp = (const float4*)(a_s + hd * 32);
  const float4* dp = (const float4*)(a_d + hd * 32);
  float ss = 0.f, sd = 0.f;
#pragma unroll
  for (int j = 0; j < 8; ++j) {
    float4 hv = hp[j], av = sp[j], bv = dp[j];
    ss += hv.x * av.x + hv.y * av.y + hv.z * av.z + hv.w * av.w;
    sd += hv.x * bv.x + hv.y * bv.y + hv.z * bv.z + hv.w * bv.w;
  }
  out_s[idx] = ss;
  out_d[idx] = sd;
}

__device__ __forceinline__ void edge_sd(const int* __restrict__ ei, int i, int E,
                                        int& s, int& d) {
  if (i < E) { s = ei[i]; d = ei[E + i]; }
  else       { s = d = i - E; }            // appended self-loops
}

// ---------- edge pass 1: segment max of leaky_relu(as[src]+ad[dst]) ----------
__global__ void edge_max_kernel(const int* __restrict__ ei,
                                const float* __restrict__ as_,
                                const float* __restrict__ ad_,
                                unsigned* __restrict__ menc,
                                int E, int Etot)
{
  int i = blockIdx.x * blockDim.x + threadIdx.x;
  if (i >= Etot) return;
  int s, d; edge_sd(ei, i, E, s, d);
#pragma unroll
  for (int h = 0; h < 4; ++h) {
    float e = as_[s * 4 + h] + ad_[d * 4 + h];
    e = (e > 0.f) ? e : NEG_SLOPE * e;
    atomicMax(&menc[d * 4 + h], mapf(e));
  }
}

// ---------- edge pass 2: denom[dst,h] += exp(e - m[dst,h]) ----------
__global__ void edge_denom_kernel(const int* __restrict__ ei,
                                  const float* __restrict__ as_,
                                  const float* __restrict__ ad_,
                                  const unsigned* __restrict__ menc,
                                  float* __restrict__ denom,
                                  int E, int Etot)
{
  int i = blockIdx.x * blockDim.x + threadIdx.x;
  if (i >= Etot) return;
  int s, d; edge_sd(ei, i, E, s, d);
#pragma unroll
  for (int h = 0; h < 4; ++h) {
    float e = as_[s * 4 + h] + ad_[d * 4 + h];
    e = (e > 0.f) ? e : NEG_SLOPE * e;
    float ex = expf(e - unmapf(menc[d * 4 + h]));
    atomicAdd(&denom[d * 4 + h], ex);
  }
}

// ---------- edge pass 3: out[dst,:] += alpha * h[src,:]  (one wave / edge) ----------
__global__ void edge_agg_kernel(const int* __restrict__ ei,
                                const float* __restrict__ as_,
                                const float* __restrict__ ad_,
                                const unsigned* __restrict__ menc,
                                const float* __restrict__ denom,
                                const float* __restrict__ hfeat,
                                float* __restrict__ out,
                                int E, int Etot)
{
  int gt   = blockIdx.x * blockDim.x + threadIdx.x;
  int w    = gt >> 5;
  int lane = gt & 31;
  if (w >= Etot) return;
  int s, d; edge_sd(ei, w, E, s, d);

  float alpha[4];
#pragma unroll
  for (int h = 0; h < 4; ++h) {
    float e = as_[s * 4 + h] + ad_[d * 4 + h];
    e = (e > 0.f) ? e : NEG_SLOPE * e;
    float ex = expf(e - unmapf(menc[d * 4 + h]));
    alpha[h] = ex / denom[d * 4 + h];
  }
  const float* hs = hfeat + (size_t)s * 128;
  float*       od = out   + (size_t)d * 128;
#pragma unroll
  for (int j = 0; j < 4; ++j) {          // channel c = lane + 32*j, head == j
    int c = lane + 32 * j;
    atomicAdd(&od[c], alpha[j] * hs[c]);
  }
}

// ---------- layer-1 epilogue: x = relu(x + b[col]) (concat heads, 128 cols) ----------
__global__ void bias_relu_128(float* __restrict__ x, const float* __restrict__ b,
                              size_t total)
{
  size_t i = (size_t)blockIdx.x * blockDim.x + threadIdx.x;
  if (i >= total) return;
  float v = x[i] + b[(int)(i & 127)];
  x[i] = fmaxf(v, 0.f);
}

// ---------- layer-2 epilogue: mean over 4 heads, +b, relu -> [N,32] ----------
__global__ void mean_bias_relu(const float* __restrict__ agg,
                               const float* __restrict__ b,
                               float* __restrict__ out, int n32)
{
  int i = blockIdx.x * blockDim.x + threadIdx.x;
  if (i >= n32) return;
  int node = i >> 5, c = i & 31;
  const float* p = agg + (size_t)node * 128;
  float v = 0.25f * (p[c] + p[c + 32] + p[c + 64] + p[c + 96]) + b[c];
  out[i] = fmaxf(v, 0.f);
}

extern "C" void kernel_launch(void* const* d_in, const int* in_sizes, int n_in,
                              void* d_out, int out_size, void* d_ws, size_t ws_size,
                              hipStream_t stream)
{
  const float* x    = (const float*)d_in[0];
  const int*   ei   = (const int*)  d_in[1];
  const float* W1   = (const float*)d_in[2];
  const float* a_s1 = (const float*)d_in[3];
  const float* a_d1 = (const float*)d_in[4];
  const float* b1   = (const float*)d_in[5];
  const float* W2   = (const float*)d_in[6];
  const float* a_s2 = (const float*)d_in[7];
  const float* a_d2 = (const float*)d_in[8];
  const float* b2   = (const float*)d_in[9];
  const float* Wc   = (const float*)d_in[10];
  const float* bc   = (const float*)d_in[11];

  const int N    = in_sizes[0] / 128;
  const int E    = in_sizes[1] / 2;
  const int Etot = E + N;

  // workspace carve-out
  char* ws = (char*)d_ws;
  size_t off = 0;
  auto alloc = [&](size_t bytes) -> void* {
    void* p = ws + off;
    off += (bytes + 255) & ~(size_t)255;
    return p;
  };
  float*    B0    = (float*)   alloc((size_t)N * 128 * 4);  // h1 / h2
  float*    B1    = (float*)   alloc((size_t)N * 128 * 4);  // agg1->feat1 / agg2
  float*    as_   = (float*)   alloc((size_t)N * 4 * 4);
  float*    ad_   = (float*)   alloc((size_t)N * 4 * 4);
  unsigned* menc  = (unsigned*)alloc((size_t)N * 4 * 4);
  float*    denom = (float*)   alloc((size_t)N * 4 * 4);
  float*    feat2 = (float*)   alloc((size_t)N * 32 * 4);
  (void)ws_size; (void)n_in; (void)out_size;

  const int mtiles = N / 16;            // 100000 % 16 == 0
  const int n4     = N * 4;
  const unsigned gE   = (unsigned)((Etot + 255) / 256);
  const unsigned gE32 = (unsigned)(((long long)Etot * 32 + 255) / 256);

  // ================= Layer 1 =================
  gemm_wmma_f32<128, 128, 64><<<dim3(mtiles, 2), 128, 0, stream>>>(x, W1, nullptr, B0);
  attn_coef<<<(n4 + 255) / 256, 256, 0, stream>>>(B0, a_s1, a_d1, as_, ad_, n4);
  (void)hipMemsetAsync(menc,  0, (size_t)N * 4 * 4,   stream);
  (void)hipMemsetAsync(denom, 0, (size_t)N * 4 * 4,   stream);
  (void)hipMemsetAsync(B1,    0, (size_t)N * 128 * 4, stream);
  edge_max_kernel  <<<gE,   256, 0, stream>>>(ei, as_, ad_, menc, E, Etot);
  edge_denom_kernel<<<gE,   256, 0, stream>>>(ei, as_, ad_, menc, denom, E, Etot);
  edge_agg_kernel  <<<gE32, 256, 0, stream>>>(ei, as_, ad_, menc, denom, B0, B1, E, Etot);
  {
    size_t tot = (size_t)N * 128;
    bias_relu_128<<<(unsigned)((tot + 255) / 256), 256, 0, stream>>>(B1, b1, tot);
  }

  // ================= Layer 2 =================
  gemm_wmma_f32<128, 128, 64><<<dim3(mtiles, 2), 128, 0, stream>>>(B1, W2, nullptr, B0);
  attn_coef<<<(n4 + 255) / 256, 256, 0, stream>>>(B0, a_s2, a_d2, as_, ad_, n4);
  (void)hipMemsetAsync(menc,  0, (size_t)N * 4 * 4,   stream);
  (void)hipMemsetAsync(denom, 0, (size_t)N * 4 * 4,   stream);
  (void)hipMemsetAsync(B1,    0, (size_t)N * 128 * 4, stream);   // feat1 fully consumed above
  edge_max_kernel  <<<gE,   256, 0, stream>>>(ei, as_, ad_, menc, E, Etot);
  edge_denom_kernel<<<gE,   256, 0, stream>>>(ei, as_, ad_, menc, denom, E, Etot);
  edge_agg_kernel  <<<gE32, 256, 0, stream>>>(ei, as_, ad_, menc, denom, B0, B1, E, Etot);
  mean_bias_relu<<<(N * 32 + 255) / 256, 256, 0, stream>>>(B1, b2, feat2, N * 32);

  // ================= Classifier =================
  gemm_wmma_f32<32, 16, 16><<<dim3(mtiles, 1), 32, 0, stream>>>(feat2, Wc, bc, (float*)d_out);
}